// TripletLossXBM_36026185679067
// MI455X (gfx1250) — compile-verified
//
#include <hip/hip_runtime.h>

// Fused triplet-loss hard-mining for gfx1250 (MI455X).
// GEMM part uses V_WMMA_F32_16X16X4_F32 (fp32, matches reference semantics).
// Shapes assumed: n % 64 == 0, m % 256 == 0, d % 4 == 0 (4096, 16384, 512).

typedef __attribute__((ext_vector_type(2))) float v2f;
typedef __attribute__((ext_vector_type(8))) float v8f;

#define TL_MARGIN 0.3f
#define TL_EPS 1e-12f

// ---------------- init: ap = bits(0.0f), an = bits(+inf) ----------------
__global__ __launch_bounds__(256)
void tl_init_kernel(unsigned* __restrict__ ap_bits, unsigned* __restrict__ an_bits, int n) {
    int i = blockIdx.x * blockDim.x + threadIdx.x;
    if (i < n) {
        ap_bits[i] = 0u;            // 0.0f  (identity for max over positive floats)
        an_bits[i] = 0x7f800000u;   // +inf  (identity for min)
    }
}

// ---------------- row squared-norms: one wave32 per row ----------------
__global__ __launch_bounds__(256)
void tl_rownorm_kernel(const float* __restrict__ src, float* __restrict__ dst,
                       int rows, int d) {
    int gw   = (blockIdx.x * blockDim.x + threadIdx.x) >> 5;
    int lane = threadIdx.x & 31;
    if (gw >= rows) return;
    const float* p = src + (size_t)gw * d;
    float s = 0.0f;
    for (int k = lane; k < d; k += 32) { float v = p[k]; s = fmaf(v, v, s); }
#pragma unroll
    for (int off = 16; off > 0; off >>= 1) s += __shfl_xor(s, off, 32);
    if (lane == 0) dst[gw] = s;
}

// ---------------- fused WMMA GEMM + masked row max/min ----------------
// Block: 256 threads = 8 waves, tile 64 rows x 256 cols.
// Wave grid 2(M) x 4(N): each wave owns a 32x64 patch = 2x4 tiles of 16x16.
__global__ __launch_bounds__(256)
void tl_fused_wmma_kernel(const float* __restrict__ X, const float* __restrict__ Y,
                          const int*   __restrict__ tc, const int*  __restrict__ tr,
                          const float* __restrict__ xx, const float* __restrict__ yy,
                          int d,
                          unsigned* __restrict__ ap_bits, unsigned* __restrict__ an_bits) {
    const int lane   = threadIdx.x & 31;
    const int w      = threadIdx.x >> 5;
    const int waveM  = w >> 2;       // 0..1
    const int waveN  = w & 3;        // 0..3
    const int lane16 = lane & 15;
    const int hihalf = lane >> 4;    // 0 or 1
    const int klo    = hihalf * 2;   // K sub-offset per ISA 32-bit A/B layout

    const int rb = blockIdx.x * 64  + waveM * 32;   // first row of this wave
    const int cb = blockIdx.y * 256 + waveN * 64;   // first col of this wave

    // A fragments: lane holds X[row][k + klo .. k + klo + 1]
    const float* pa0 = X + (size_t)(rb      + lane16) * d + klo;
    const float* pa1 = X + (size_t)(rb + 16 + lane16) * d + klo;
    // B fragments: lane holds Y[col][k + klo .. k + klo + 1]  (B[k][n] = Y[n][k])
    const float* pb0 = Y + (size_t)(cb      + lane16) * d + klo;
    const float* pb1 = Y + (size_t)(cb + 16 + lane16) * d + klo;
    const float* pb2 = Y + (size_t)(cb + 32 + lane16) * d + klo;
    const float* pb3 = Y + (size_t)(cb + 48 + lane16) * d + klo;

    v8f acc[2][4];
#pragma unroll
    for (int i = 0; i < 2; ++i)
#pragma unroll
        for (int j = 0; j < 4; ++j) acc[i][j] = (v8f){};

#pragma unroll 2
    for (int k = 0; k < d; k += 4) {
        v2f a0 = *(const v2f*)(pa0 + k);
        v2f a1 = *(const v2f*)(pa1 + k);
        v2f b0 = *(const v2f*)(pb0 + k);
        v2f b1 = *(const v2f*)(pb1 + k);
        v2f b2 = *(const v2f*)(pb2 + k);
        v2f b3 = *(const v2f*)(pb3 + k);
        acc[0][0] = __builtin_amdgcn_wmma_f32_16x16x4_f32(false, a0, false, b0, (short)0, acc[0][0], false, false);
        acc[0][1] = __builtin_amdgcn_wmma_f32_16x16x4_f32(false, a0, false, b1, (short)0, acc[0][1], false, false);
        acc[0][2] = __builtin_amdgcn_wmma_f32_16x16x4_f32(false, a0, false, b2, (short)0, acc[0][2], false, false);
        acc[0][3] = __builtin_amdgcn_wmma_f32_16x16x4_f32(false, a0, false, b3, (short)0, acc[0][3], false, false);
        acc[1][0] = __builtin_amdgcn_wmma_f32_16x16x4_f32(false, a1, false, b0, (short)0, acc[1][0], false, false);
        acc[1][1] = __builtin_amdgcn_wmma_f32_16x16x4_f32(false, a1, false, b1, (short)0, acc[1][1], false, false);
        acc[1][2] = __builtin_amdgcn_wmma_f32_16x16x4_f32(false, a1, false, b2, (short)0, acc[1][2], false, false);
        acc[1][3] = __builtin_amdgcn_wmma_f32_16x16x4_f32(false, a1, false, b3, (short)0, acc[1][3], false, false);
    }

    const float INF = __uint_as_float(0x7f800000u);

    // Epilogue. C/D layout: VGPR e, lanes 0-15 -> M=e, N=lane; lanes 16-31 -> M=e+8.
#pragma unroll
    for (int mt = 0; mt < 2; ++mt) {
        // per-lane row data (row depends on e and lane half only)
        float xv[8]; int tcv[8];
#pragma unroll
        for (int e = 0; e < 8; ++e) {
            int row = rb + mt * 16 + 8 * hihalf + e;
            xv[e]  = xx[row];
            tcv[e] = tc[row];
        }
        float apc[8], anc[8];
#pragma unroll
        for (int e = 0; e < 8; ++e) { apc[e] = -INF; anc[e] = INF; }

#pragma unroll
        for (int nt = 0; nt < 4; ++nt) {
            int   col = cb + nt * 16 + lane16;
            float yv  = yy[col];
            int   trv = tr[col];
#pragma unroll
            for (int e = 0; e < 8; ++e) {
                float s    = acc[mt][nt][e];
                float d2   = fmaf(-2.0f, s, xv[e] + yv);
                float dist = sqrtf(fmaxf(d2, TL_EPS));
                if (tcv[e] == trv) apc[e] = fmaxf(apc[e], dist);
                else               anc[e] = fminf(anc[e], dist);
            }
        }

        // reduce over the 16 lanes holding the same row (xor masks stay in half)
#pragma unroll
        for (int e = 0; e < 8; ++e) {
            float ap = apc[e], an = anc[e];
#pragma unroll
            for (int off = 1; off < 16; off <<= 1) {
                ap = fmaxf(ap, __shfl_xor(ap, off, 32));
                an = fminf(an, __shfl_xor(an, off, 32));
            }
            if (lane16 == 0) {
                int row = rb + mt * 16 + 8 * hihalf + e;
                // positive floats: signed-int compare == float compare
                if (ap > 0.0f) atomicMax((int*)&ap_bits[row], __float_as_int(ap));
                if (an <  INF) atomicMin((int*)&an_bits[row], __float_as_int(an));
            }
        }
    }
}

// ---------------- deterministic single-block mean ----------------
__global__ __launch_bounds__(256)
void tl_finalize_kernel(const unsigned* __restrict__ ap_bits,
                        const unsigned* __restrict__ an_bits,
                        float* __restrict__ out, int n) {
    __shared__ float sdata[256];
    float s = 0.0f;
    for (int i = threadIdx.x; i < n; i += 256) {
        unsigned ab = ap_bits[i], nb = an_bits[i];
        bool has_both = (ab != 0u) && (nb != 0x7f800000u);
        float ap = has_both ? __uint_as_float(ab) : 0.0f;
        float an = has_both ? __uint_as_float(nb) : 0.0f;
        s += fmaxf(ap - an + TL_MARGIN, 0.0f);
    }
    sdata[threadIdx.x] = s;
    __syncthreads();
#pragma unroll
    for (int stride = 128; stride > 0; stride >>= 1) {
        if ((int)threadIdx.x < stride) sdata[threadIdx.x] += sdata[threadIdx.x + stride];
        __syncthreads();
    }
    if (threadIdx.x == 0) out[0] = sdata[0] / (float)n;
}

extern "C" void kernel_launch(void* const* d_in, const int* in_sizes, int n_in,
                              void* d_out, int out_size, void* d_ws, size_t ws_size,
                              hipStream_t stream) {
    const float* X  = (const float*)d_in[0];   // inputs_col [n, d]
    const int*   tc = (const int*)  d_in[1];   // targets_col [n]
    const float* Y  = (const float*)d_in[2];   // inputs_row [m, d]
    const int*   tr = (const int*)  d_in[3];   // targets_row [m]
    const int n = in_sizes[1];
    const int m = in_sizes[3];
    const int d = in_sizes[0] / n;
    float* out = (float*)d_out;

    // workspace layout: ap_bits[n] | an_bits[n] | xx[n] | yy[m]
    unsigned* ap_bits = (unsigned*)d_ws;
    unsigned* an_bits = ap_bits + n;
    float*    xx      = (float*)(an_bits + n);
    float*    yy      = xx + n;

    tl_init_kernel<<<(n + 255) / 256, 256, 0, stream>>>(ap_bits, an_bits, n);

    // one wave per row -> 8 rows per 256-thread block
    tl_rownorm_kernel<<<(n + 7) / 8, 256, 0, stream>>>(X, xx, n, d);
    tl_rownorm_kernel<<<(m + 7) / 8, 256, 0, stream>>>(Y, yy, m, d);

    dim3 grid(n / 64, m / 256);
    tl_fused_wmma_kernel<<<grid, 256, 0, stream>>>(X, Y, tc, tr, xx, yy, d,
                                                   ap_bits, an_bits);

    tl_finalize_kernel<<<1, 256, 0, stream>>>(ap_bits, an_bits, out, n);
}